// MS_SSIMLoss_66795331388108
// MI455X (gfx1250) — compile-verified
//
#include <hip/hip_runtime.h>
#include <math.h>

// ---------------------------------------------------------------------------
// MS-SSIM loss for MI455X (gfx1250, wave32). The 11x11 separable Gaussian
// convolutions (the FLOP-dominant part) are executed as banded-matrix matmuls
// on V_WMMA_F32_16X16X4_F32. One wave computes one 16x16 output tile:
//   horizontal: T[26x16] = Q[26x28] x Gh[28x16]   (Gh[j][c] = g[j-c], banded)
//   vertical:   S[16x16] = Gv[16x28] x T[28x16]   (Gv[r][i] = g[i-r], banded)
// K padded 26->28; padded rows/cols are exact zeros so the algebra is exact.
// Boundary handling is branchless (clamped address + multiplicative mask) to
// avoid exec-mask diamonds around every load.
// ---------------------------------------------------------------------------

typedef float v2f __attribute__((ext_vector_type(2)));
typedef float v8f __attribute__((ext_vector_type(8)));

#define MSS_C1 1.0e-4f   // (0.01 * val_range)^2
#define MSS_C2 9.0e-4f   // (0.03 * val_range)^2

// Normalized gaussian(11, sigma=1.5)
__constant__ float c_gw[11] = {
  0.00102840f, 0.00759875f, 0.03600077f, 0.10936080f, 0.21300554f, 0.26601172f,
  0.21300554f, 0.10936080f, 0.03600077f, 0.00759875f, 0.00102840f
};

__device__ __forceinline__ float gband(int j, int r) {
  unsigned d = (unsigned)(j - r);
  return (d <= 10u) ? c_gw[d] : 0.0f;
}

__device__ __forceinline__ v8f wmma_f32(v2f a, v2f b, v8f c) {
  // (neg_a, A, neg_b, B, c_mod, C, reuse_a, reuse_b)
  return __builtin_amdgcn_wmma_f32_16x16x4_f32(false, a, false, b, (short)0, c,
                                               false, false);
}

// ---------------------------------------------------------------------------
// Kernel 1: per-pixel channel softmax (C=12) + one-hot of target.
// ---------------------------------------------------------------------------
__global__ void msssim_softmax_onehot(const float* __restrict__ pred,
                                      const int* __restrict__ target,
                                      float* __restrict__ X,
                                      float* __restrict__ Y,
                                      int B, int HW)
{
  int idx = blockIdx.x * blockDim.x + threadIdx.x;
  if (idx >= B * HW) return;
  int b = idx / HW, hw = idx - b * HW;
  const float* p = pred + (size_t)b * 12 * HW + hw;
  float v[12];
  float mx = -3.4e38f;
#pragma unroll
  for (int c = 0; c < 12; ++c) { v[c] = p[(size_t)c * HW]; mx = fmaxf(mx, v[c]); }
  float s = 0.f;
#pragma unroll
  for (int c = 0; c < 12; ++c) { v[c] = __expf(v[c] - mx); s += v[c]; }
  float inv = 1.f / s;
  int t = target[idx];
  float* xo = X + (size_t)b * 12 * HW + hw;
  float* yo = Y + (size_t)b * 12 * HW + hw;
#pragma unroll
  for (int c = 0; c < 12; ++c) {
    xo[(size_t)c * HW] = v[c] * inv;
    yo[(size_t)c * HW] = (c == t) ? 1.f : 0.f;
  }
}

// ---------------------------------------------------------------------------
// Kernel 2: zero the per-(scale,image) accumulators (deterministic per call).
// ---------------------------------------------------------------------------
__global__ void msssim_zero(float* __restrict__ p, int n)
{
  int i = blockIdx.x * blockDim.x + threadIdx.x;
  if (i < n) p[i] = 0.f;
}

// ---------------------------------------------------------------------------
// Kernel 3: SSIM over one scale. One 32-thread block (== one wave) computes a
// 16x16 tile of the VALID conv output for one (b,c) image via WMMA, forms the
// cs/ss maps, and atomically accumulates masked spatial sums.
// ---------------------------------------------------------------------------
__global__ void __launch_bounds__(32)
msssim_ssim_tiles(const float* __restrict__ X, const float* __restrict__ Y,
                  float* __restrict__ ssAcc, float* __restrict__ csAcc,
                  int Hs, int Ws)
{
  const int lane = threadIdx.x;
  const int n    = lane & 15;   // N column (B/C/D) and A/M row index
  const int half = lane >> 4;   // 0: lanes 0-15, 1: lanes 16-31
  const int Ho = Hs - 10, Wo = Ws - 10;
  const int th0 = blockIdx.y * 16;
  const int tw0 = blockIdx.x * 16;
  const int img = blockIdx.z;
  const float* xim = X + (size_t)img * Hs * Ws;
  const float* yim = Y + (size_t)img * Hs * Ws;

  __shared__ float ldsT[5][32][16];   // T re-striping buffer (5 quantities)

  // Gaussian band fragments for the 7 K-steps. These serve as BOTH the
  // horizontal B operand (Gh[j][n] = g[j-n]) and the vertical A operand
  // (Gv[n][k] = g[k-n]) -- the same banded values. Hoisted once per wave.
  v2f gfrag[7];
#pragma unroll
  for (int t = 0; t < 7; ++t) {
    int j0 = 4 * t + 2 * half;
    gfrag[t].x = gband(j0, n);
    gfrag[t].y = gband(j0 + 1, n);
  }

  const v8f vzero = {0.f, 0.f, 0.f, 0.f, 0.f, 0.f, 0.f, 0.f};
  v8f T[5][2];
#pragma unroll
  for (int q = 0; q < 5; ++q) {
    T[q][0] = vzero;
    T[q][1] = vzero;
  }

  // ---- horizontal pass: T[q] = Q[q] x Gh ---------------------------------
#pragma unroll
  for (int chunk = 0; chunk < 2; ++chunk) {
    const int   i     = chunk * 16 + n;        // Q row this lane supplies to A
    const int   gr    = th0 + i;               // global image row
    const float rmask = ((i < 26) && (gr < Hs)) ? 1.f : 0.f;
    const int   grc   = min(gr, Hs - 1);       // clamped (always loadable)
    const float* xrow = xim + (size_t)grc * Ws;
    const float* yrow = yim + (size_t)grc * Ws;
#pragma unroll
    for (int t = 0; t < 7; ++t) {
      const int j0 = 4 * t + 2 * half;         // this lane's K columns: j0, j0+1
      const int c0 = tw0 + j0;
      // branchless clamp-and-mask boundary handling
      const float m0 = rmask * (((j0 < 26) && (c0 < Ws)) ? 1.f : 0.f);
      const float m1 = rmask * (((j0 + 1 < 26) && (c0 + 1 < Ws)) ? 1.f : 0.f);
      const int   c0c = min(c0, Ws - 1);
      const int   c1c = min(c0 + 1, Ws - 1);
      const float x0 = xrow[c0c] * m0;
      const float x1 = xrow[c1c] * m1;
      const float y0 = yrow[c0c] * m0;
      const float y1 = yrow[c1c] * m1;
      const v2f bg = gfrag[t];
      v2f a;
      a.x = x0;      a.y = x1;      T[0][chunk] = wmma_f32(a, bg, T[0][chunk]);
      a.x = y0;      a.y = y1;      T[1][chunk] = wmma_f32(a, bg, T[1][chunk]);
      a.x = x0 * x0; a.y = x1 * x1; T[2][chunk] = wmma_f32(a, bg, T[2][chunk]);
      a.x = y0 * y0; a.y = y1 * y1; T[3][chunk] = wmma_f32(a, bg, T[3][chunk]);
      a.x = x0 * y0; a.y = x1 * y1; T[4][chunk] = wmma_f32(a, bg, T[4][chunk]);
    }
  }

  // ---- re-stripe T through LDS (C/D layout -> row major) -----------------
#pragma unroll
  for (int q = 0; q < 5; ++q)
#pragma unroll
    for (int chunk = 0; chunk < 2; ++chunk)
#pragma unroll
      for (int v = 0; v < 8; ++v)
        ldsT[q][chunk * 16 + v + 8 * half][n] = T[q][chunk][v];
  __syncthreads();   // block == one wave: cheap, orders LDS write->read

  // ---- vertical pass: S[q] = Gv x T[q] -----------------------------------
  v8f S[5];
#pragma unroll
  for (int q = 0; q < 5; ++q) {
    v8f acc = vzero;
#pragma unroll
    for (int t = 0; t < 7; ++t) {
      const int k0 = 4 * t + 2 * half;
      v2f b;                       // B: T rows k0, k0+1 at column n
      b.x = ldsT[q][k0][n];
      b.y = ldsT[q][k0 + 1][n];
      acc = wmma_f32(gfrag[t], b, acc);
    }
    S[q] = acc;
  }

  // ---- SSIM maps + masked tile reduction ---------------------------------
  float ssp = 0.f, csp = 0.f;
#pragma unroll
  for (int v = 0; v < 8; ++v) {
    int ho = th0 + v + 8 * half;
    int wo = tw0 + n;
    float mask = ((ho < Ho) && (wo < Wo)) ? 1.f : 0.f;
    float a1  = S[0][v], a2 = S[1][v];
    float a1s = a1 * a1, a2s = a2 * a2, a12 = a1 * a2;
    float v1  = S[2][v] - a1s;
    float v2  = S[3][v] - a2s;
    float cov = S[4][v] - a12;
    float cs = (2.f * cov + MSS_C2) / (v1 + v2 + MSS_C2);
    float ss = (2.f * a12 + MSS_C1) / (a1s + a2s + MSS_C1) * cs;
    csp += cs * mask;
    ssp += ss * mask;
  }
#pragma unroll
  for (int off = 16; off > 0; off >>= 1) {
    ssp += __shfl_xor(ssp, off, 32);
    csp += __shfl_xor(csp, off, 32);
  }
  if (lane == 0) {
    atomicAdd(&ssAcc[img], ssp);
    atomicAdd(&csAcc[img], csp);
  }
}

// ---------------------------------------------------------------------------
// Kernel 4: 2x2 average pooling (between scales).
// ---------------------------------------------------------------------------
__global__ void msssim_avgpool2(const float* __restrict__ in,
                                float* __restrict__ outp,
                                int Hn, int Wn, int nimg)
{
  int idx = blockIdx.x * blockDim.x + threadIdx.x;
  int total = nimg * Hn * Wn;
  if (idx >= total) return;
  int wo = idx % Wn;
  int t  = idx / Wn;
  int ho = t % Hn;
  int im = t / Hn;
  int Wi = 2 * Wn;
  const float* ip = in + (((size_t)im * 2 * Hn) + 2 * ho) * Wi + 2 * wo;
  outp[idx] = 0.25f * (ip[0] + ip[1] + ip[Wi] + ip[Wi + 1]);
}

// ---------------------------------------------------------------------------
// Kernel 5: finalize — means, clamp, weighted geometric product, loss.
// ---------------------------------------------------------------------------
__global__ void msssim_finalize(const float* __restrict__ ssAcc,
                                const float* __restrict__ csAcc,
                                float* __restrict__ out)
{
  __shared__ float sm[48];
  const float w[5]  = {0.0448f, 0.2856f, 0.3001f, 0.2363f, 0.1333f};
  const int   Hs[5] = {512, 256, 128, 64, 32};
  int t = threadIdx.x;
  if (t < 48) {
    float msss = 1.f;
#pragma unroll
    for (int s = 0; s < 5; ++s) {
      int Ho = Hs[s] - 10;
      float denom = (float)Ho * (float)Ho;
      float v = ((s < 4) ? csAcc[s * 48 + t] : ssAcc[s * 48 + t]) / denom;
      v = fmaxf(v, 0.f);
      msss *= powf(v, w[s]);
    }
    sm[t] = msss;
  }
  __syncthreads();
  if (t == 0) {
    float a = 0.f;
    for (int i = 0; i < 48; ++i) a += sm[i];
    out[0] = 1.f - a / 48.f;
  }
}

// ---------------------------------------------------------------------------
// Launcher
// ---------------------------------------------------------------------------
extern "C" void kernel_launch(void* const* d_in, const int* in_sizes, int n_in,
                              void* d_out, int out_size, void* d_ws, size_t ws_size,
                              hipStream_t stream)
{
  (void)in_sizes; (void)n_in; (void)out_size; (void)ws_size;
  const float* pred   = (const float*)d_in[0];
  const int*   target = (const int*)d_in[1];
  float*       out    = (float*)d_out;
  float*       ws     = (float*)d_ws;

  const int B = 4, NIMG = 48;                 // B*C images
  const int Hs[5] = {512, 256, 128, 64, 32};

  // Workspace layout (floats): X pyramid, Y pyramid, accumulators.
  // Total ~= 33.5M floats (~134 MB) + 480.
  float* X[5];
  float* Y[5];
  size_t off = 0;
  for (int s = 0; s < 5; ++s) { X[s] = ws + off; off += (size_t)NIMG * Hs[s] * Hs[s]; }
  for (int s = 0; s < 5; ++s) { Y[s] = ws + off; off += (size_t)NIMG * Hs[s] * Hs[s]; }
  float* ssAcc = ws + off; off += 5 * NIMG;
  float* csAcc = ws + off; off += 5 * NIMG;

  const int HW = 512 * 512;
  int npix = B * HW;
  msssim_softmax_onehot<<<(npix + 255) / 256, 256, 0, stream>>>(
      pred, target, X[0], Y[0], B, HW);
  msssim_zero<<<1, 512, 0, stream>>>(ssAcc, 2 * 5 * NIMG);

  for (int s = 0; s < 5; ++s) {
    int H  = Hs[s];
    int Ho = H - 10;
    int tiles = (Ho + 15) / 16;
    dim3 grid(tiles, tiles, NIMG);
    msssim_ssim_tiles<<<grid, 32, 0, stream>>>(
        X[s], Y[s], ssAcc + s * NIMG, csAcc + s * NIMG, H, H);
    if (s < 4) {
      int Hn = Hs[s + 1];
      int total = NIMG * Hn * Hn;
      msssim_avgpool2<<<(total + 255) / 256, 256, 0, stream>>>(
          X[s], X[s + 1], Hn, Hn, NIMG);
      msssim_avgpool2<<<(total + 255) / 256, 256, 0, stream>>>(
          Y[s], Y[s + 1], Hn, Hn, NIMG);
    }
  }

  msssim_finalize<<<1, 64, 0, stream>>>(ssAcc, csAcc, out);
}